// MyRNN_38311108280354
// MI455X (gfx1250) — compile-verified
//
#include <hip/hip_runtime.h>

// RNN: h_t = tanh(h_{t-1} @ Wh^T + x_t @ Wx^T); o_t = h_t @ Wo^T
// B=256, L=1024, D=H=O=128, fp32.
//
// 3-phase plan driven by MI455X roofline reasoning:
//   K1 (full GPU, HBM-bound):  out <- xs @ Wx^T        (xproj staged in `out`)
//   K2 (serial scan, 16 blks): out[.,l,:] <- h_l, where
//        h_l = tanh(h_{l-1} @ Wh^T + xproj_l); xproj frag is the WMMA C operand,
//        loaded one step ahead; only 32 WMMA/wave on the critical path.
//   K3 (full GPU, HBM-bound):  out <- out @ Wo^T in place (row-local GEMM).
// fp32 WMMA (16x16x4) keeps reference precision; weight B-fragments live in
// VGPRs; hidden state in LDS with padded stride (conflict-free ds_load_b64).

typedef __attribute__((ext_vector_type(2))) float v2f;
typedef __attribute__((ext_vector_type(8))) float v8f;

#define BB 256
#define LL 1024
#define DD 128
#define HH 128
#define OO 128
#define HS 132  // padded LDS row stride (floats): lane r -> bank 4r

__device__ __forceinline__ float fast_tanh(float x) {
#if __has_builtin(__builtin_amdgcn_tanhf)
  return __builtin_amdgcn_tanhf(x);   // native V_TANH_F32 (TRANS unit)
#else
  return tanhf(x);
#endif
}

// ---------------- K1: out = xs @ Wx^T  over flattened [B*L, 128] ----------------
__global__ __launch_bounds__(256) void xproj_kernel(const float* __restrict__ xs,
                                                    const float* __restrict__ Wx,
                                                    float* __restrict__ xp) {
  const int tid = threadIdx.x;
  const int wave = tid >> 5, lane = tid & 31;
  const int r = lane & 15, hi = lane >> 4, colsel = hi * 2;
  const size_t mbase = (size_t)blockIdx.x * 16;

  // B-frag: B[k][n] = Wx[wave*16+n][k]  -> per-lane float2 at [wrow][4k+colsel]
  const int wrow = wave * 16 + r;
  v2f bfrag[32];
#pragma unroll
  for (int k = 0; k < 32; ++k)
    bfrag[k] = *(const v2f*)(Wx + wrow * DD + 4 * k + colsel);

  const float* arow = xs + (mbase + r) * DD + colsel;
  v8f acc = {};
#pragma unroll
  for (int k = 0; k < 32; ++k) {
    v2f a = *(const v2f*)(arow + 4 * k);
    acc = __builtin_amdgcn_wmma_f32_16x16x4_f32(
        false, a, false, bfrag[k], (short)0, acc, false, false);
  }
  // D frag: VGPR v -> row mbase + v + 8*hi, col wave*16 + r
  float* orow = xp + (mbase + 8 * hi) * OO + wave * 16 + r;
#pragma unroll
  for (int v = 0; v < 8; ++v) orow[(size_t)v * OO] = acc[v];
}

// ---------------- K2: sequential scan over L, h staged in LDS ----------------
__global__ __launch_bounds__(256) void scan_kernel(const float* __restrict__ Wh,
                                                   float* __restrict__ io) {
  __shared__ float hbuf[16 * HS];
  const int tid = threadIdx.x;
  const int wave = tid >> 5, lane = tid & 31;
  const int r = lane & 15, hi = lane >> 4, colsel = hi * 2;
  const int bbase = blockIdx.x * 16;

  const int wrow = wave * 16 + r;
  v2f whB[32];
#pragma unroll
  for (int k = 0; k < 32; ++k)
    whB[k] = *(const v2f*)(Wh + wrow * HH + 4 * k + colsel);

  for (int i = tid; i < 16 * HS; i += 256) hbuf[i] = 0.0f;
  __syncthreads();

  // C/D fragment location in io for this wave: element (v,l) of batch row
  // (bbase + v + 8*hi), hidden col wave*16 + r.
  float* crow = io + (size_t)(bbase + 8 * hi) * LL * OO + wave * 16 + r;

  float xp[8];
#pragma unroll
  for (int v = 0; v < 8; ++v) xp[v] = crow[(size_t)v * LL * OO];  // l = 0

  for (int l = 0; l < LL; ++l) {
    // Software-prefetch next step's xproj fragment (uniform, clamped).
    const int ln = (l + 1 < LL) ? (l + 1) : l;
    float xpn[8];
#pragma unroll
    for (int v = 0; v < 8; ++v)
      xpn[v] = crow[((size_t)v * LL + ln) * OO];

    // acc starts from xproj (C operand); two independent WMMA chains.
    v8f acc = {xp[0], xp[1], xp[2], xp[3], xp[4], xp[5], xp[6], xp[7]};
    v8f acc1 = {};
#pragma unroll
    for (int k = 0; k < 16; ++k) {
      v2f a0 = *(const v2f*)(&hbuf[r * HS + 8 * k + colsel]);
      v2f a1 = *(const v2f*)(&hbuf[r * HS + 8 * k + 4 + colsel]);
      acc = __builtin_amdgcn_wmma_f32_16x16x4_f32(
          false, a0, false, whB[2 * k], (short)0, acc, false, false);
      acc1 = __builtin_amdgcn_wmma_f32_16x16x4_f32(
          false, a1, false, whB[2 * k + 1], (short)0, acc1, false, false);
    }

    float hv[8];
#pragma unroll
    for (int i = 0; i < 8; ++i) hv[i] = fast_tanh(acc[i] + acc1[i]);

    __syncthreads();  // all waves finished reading h_{l-1}
#pragma unroll
    for (int v = 0; v < 8; ++v) hbuf[(v + 8 * hi) * HS + wave * 16 + r] = hv[v];
    // Publish h_l to global, overwriting its own xproj slot.
#pragma unroll
    for (int v = 0; v < 8; ++v) crow[((size_t)v * LL + l) * OO] = hv[v];
    __syncthreads();  // h_l visible in LDS for next step

#pragma unroll
    for (int v = 0; v < 8; ++v) xp[v] = xpn[v];
  }
}

// ---------------- K3: in-place out = out @ Wo^T over [B*L, 128] ----------------
__global__ __launch_bounds__(256) void outproj_kernel(const float* __restrict__ Wo,
                                                      float* io) {
  const int tid = threadIdx.x;
  const int wave = tid >> 5, lane = tid & 31;
  const int r = lane & 15, hi = lane >> 4, colsel = hi * 2;
  const size_t mbase = (size_t)blockIdx.x * 16;

  const int wrow = wave * 16 + r;
  v2f woB[32];
#pragma unroll
  for (int k = 0; k < 32; ++k)
    woB[k] = *(const v2f*)(Wo + wrow * HH + 4 * k + colsel);

  const float* arow = io + (mbase + r) * HH + colsel;
  v8f acc = {};
#pragma unroll
  for (int k = 0; k < 32; ++k) {
    v2f a = *(const v2f*)(arow + 4 * k);  // consumed by WMMA -> load completes
    acc = __builtin_amdgcn_wmma_f32_16x16x4_f32(
        false, a, false, woB[k], (short)0, acc, false, false);
  }
  __syncthreads();  // every wave's h-tile reads are done before overwrite
  float* orow = io + (mbase + 8 * hi) * OO + wave * 16 + r;
#pragma unroll
  for (int v = 0; v < 8; ++v) orow[(size_t)v * OO] = acc[v];
}

extern "C" void kernel_launch(void* const* d_in, const int* in_sizes, int n_in,
                              void* d_out, int out_size, void* d_ws, size_t ws_size,
                              hipStream_t stream) {
  (void)in_sizes; (void)n_in; (void)d_ws; (void)ws_size; (void)out_size;
  const float* xs = (const float*)d_in[0];
  const float* Wh = (const float*)d_in[1];
  const float* Wx = (const float*)d_in[2];
  const float* Wo = (const float*)d_in[3];
  float* out = (float*)d_out;

  xproj_kernel<<<(BB * LL) / 16, 256, 0, stream>>>(xs, Wx, out);
  scan_kernel<<<BB / 16, 256, 0, stream>>>(Wh, out);
  outproj_kernel<<<(BB * LL) / 16, 256, 0, stream>>>(Wo, out);
}